// ProjectedGaussianRasterizer_68702296867457
// MI455X (gfx1250) — compile-verified
//
#include <hip/hip_runtime.h>

typedef float v2f __attribute__((ext_vector_type(2)));
typedef float v4f __attribute__((ext_vector_type(4)));
typedef float v8f __attribute__((ext_vector_type(8)));

#define GTOT 8192
#define NB   2
#define IMG  256
#define NCH  128   // 8192 / 64
#define CG   64    // gaussians staged per chunk

#define L2E 1.4426950408889634f   // log2(e)

__device__ __forceinline__ float fmin1(float a, float b) {
  return __builtin_amdgcn_fmed3f(a, b, -__builtin_inff());
}
__device__ __forceinline__ float fmax1(float a, float b) {
  return __builtin_amdgcn_fmed3f(a, b, __builtin_inff());
}

// ---------------------------------------------------------------------------
// Pass 1: stable depth rank (O(G^2), 134M compares total -- negligible).
// order[b][rank] = original index, identical to stable argsort.
// ---------------------------------------------------------------------------
__global__ __launch_bounds__(256) void gs_sort_rank(
    const float* __restrict__ depths, int* __restrict__ order) {
  __shared__ float sd[256];
  const int b = blockIdx.y;
  const int i = blockIdx.x * 256 + threadIdx.x;
  const float di = depths[b * GTOT + i];
  int rank = 0;
  for (int base = 0; base < GTOT; base += 256) {
    __syncthreads();
    sd[threadIdx.x] = depths[b * GTOT + base + threadIdx.x];
    __syncthreads();
    for (int j = 0; j < 256; ++j) {
      const float dj = sd[j];
      const int jj = base + j;
      rank += (dj < di || (dj == di && jj < i)) ? 1 : 0;
    }
  }
  order[b * GTOT + rank] = i;
}

// ---------------------------------------------------------------------------
// Pass 2: rasterize. One 16x16 pixel tile per workgroup (256 thr = 8 waves).
// Each wave owns 2 rows of 16 pixels. Per 16-gaussian step, the log2-domain
// exponent q(g,p) = log2e*power + log2(op) is computed with two
// v_wmma_f32_16x16x4_f32 (M=gaussians, N=pixels, K=6, log2 scales folded into
// the per-gaussian coefficients). alpha = exp2(min(q, log2(op))).
// WMMA output doubles as an exact cull: if max(acc) < -8 wave-wide, every
// alpha is < 1/255 (zeroed by the reference) and the step is skipped.
// Lane L(<16)/L+16 hold gaussians 0-7 / 8-15 of pixel column L; serial
// front-to-back compositing per half, stitched with one shfl_xor(16).
// LDS chunk staging is double-buffered with register preloading.
// ---------------------------------------------------------------------------
__global__ __launch_bounds__(256) void gs_raster(
    const float* __restrict__ means2d, const float* __restrict__ conics,
    const float* __restrict__ colors,  const float* __restrict__ opac,
    const int*   __restrict__ order,   float* __restrict__ out) {
  __shared__ float sf[2][CG][6];   // log2-scaled quadratic-form coefficients
  __shared__ v4f   s4[2][CG];      // {log2(op), r, g, b} per gaussian

  const int b    = blockIdx.z;
  const int tid  = threadIdx.x;
  const int lane = tid & 31;
  const int wav  = tid >> 5;
  const int colN = lane & 15;   // pixel column / WMMA N index
  const int seg  = lane >> 4;   // 0: holds K0,K1 (A) & gaussians 0-7 (D)
                                // 1: holds K2,K3 (A) & gaussians 8-15 (D)

  const float cx = blockIdx.x * 16 + 8.0f;   // tile center (pixel centers +0.5)
  const float cy = blockIdx.y * 16 + 8.0f;
  const float pxp = (float)colN - 7.5f;      // tile-local pixel center x

  // Pixel-feature B matrices (invariant over all gaussian chunks).
  // Feature order K: [px'^2, py'^2, px'*py', px', | py', 1, 0, 0]
  // B1 rows K=6,7 (seg==1 lanes) are zero => A1 seg==1 lanes may hold
  // arbitrary finite data (0 * finite == 0), enabling unconditional A loads.
  v2f B0[2], B1[2];
  for (int r = 0; r < 2; ++r) {
    const float pyp = (float)(2 * wav + r) - 7.5f;
    if (seg == 0) { B0[r] = (v2f){pxp * pxp, pyp * pyp}; B1[r] = (v2f){pyp, 1.0f}; }
    else          { B0[r] = (v2f){pxp * pyp, pxp};       B1[r] = (v2f){0.0f, 0.0f}; }
  }

  float T[2]  = {1.0f, 1.0f};
  float ir[2] = {0.0f, 0.0f}, ig[2] = {0.0f, 0.0f}, ib[2] = {0.0f, 0.0f};

  // ---- register preload of raw gaussian data (threads 0..63 only) ----
  float r_mx = 0.f, r_my = 0.f, r_ca = 0.f, r_cb = 0.f, r_cc = 0.f;
  float r_op = 1.f, r_c0 = 0.f, r_c1 = 0.f, r_c2 = 0.f;
  #define PRELOAD(CH)                                                        \
    if (tid < CG && (CH) < NCH) {                                            \
      const int idx = order[b * GTOT + (CH) * CG + tid];                     \
      r_mx = means2d[(b * GTOT + idx) * 2 + 0];                              \
      r_my = means2d[(b * GTOT + idx) * 2 + 1];                              \
      r_ca = conics[(b * GTOT + idx) * 3 + 0];                               \
      r_cb = conics[(b * GTOT + idx) * 3 + 1];                               \
      r_cc = conics[(b * GTOT + idx) * 3 + 2];                               \
      r_op = opac[b * GTOT + idx];                                           \
      r_c0 = colors[(b * GTOT + idx) * 3 + 0];                               \
      r_c1 = colors[(b * GTOT + idx) * 3 + 1];                               \
      r_c2 = colors[(b * GTOT + idx) * 3 + 2];                               \
    }

  PRELOAD(0);

  for (int ch = 0; ch < NCH; ++ch) {
    const int buf = ch & 1;
    if (tid < CG) {  // expand quadratic form in log2 domain, stage to LDS
      const float mx = r_mx - cx;
      const float my = r_my - cy;
      const float l2op = __builtin_amdgcn_logf(r_op);  // v_log_f32 == log2
      float* fp = &sf[buf][tid][0];
      *(v2f*)(fp + 0) = (v2f){-0.5f * L2E * r_ca, -0.5f * L2E * r_cc};
      *(v2f*)(fp + 2) = (v2f){-L2E * r_cb, L2E * (r_ca * mx + r_cb * my)};
      *(v2f*)(fp + 4) = (v2f){L2E * (r_cc * my + r_cb * mx),
          l2op - L2E * (0.5f * (r_ca * mx * mx + r_cc * my * my) +
                        r_cb * mx * my)};
      s4[buf][tid] = (v4f){l2op, r_c0, r_c1, r_c2};
    }
    PRELOAD(ch + 1);   // hide gather latency under this chunk's compute
    __syncthreads();

    for (int s = 0; s < 4; ++s) {          // 4 x 16 gaussians per chunk
      const int gA = s * 16 + colN;        // A-matrix row M = gaussian
      const float* fb = &sf[buf][gA][0];
      const v2f A0 = *(const v2f*)(fb + 2 * seg);  // K01 (seg0) / K23 (seg1)
      const v2f A1 = *(const v2f*)(fb + 4);        // K45; seg1 hits zero B rows
      const int gC = s * 16 + seg * 8;     // this lane's 8 D-rows (gaussians)

      for (int r = 0; r < 2; ++r) {
        v8f acc = {};
        acc = __builtin_amdgcn_wmma_f32_16x16x4_f32(false, A1, false, B1[r],
                                                    (short)0, acc, false, false);
        acc = __builtin_amdgcn_wmma_f32_16x16x4_f32(false, A0, false, B0[r],
                                                    (short)0, acc, false, false);

        // exact wave-wide cull: q <= acc, and acc < -8 => alpha < 1/255 => 0
        float m01 = fmax1(acc[0], acc[1]), m23 = fmax1(acc[2], acc[3]);
        float m45 = fmax1(acc[4], acc[5]), m67 = fmax1(acc[6], acc[7]);
        const float mx8 = fmax1(fmax1(m01, m23), fmax1(m45, m67));
        if (__any(mx8 >= -8.0f)) {
          // serial front-to-back over this lane's 8 gaussians
          const float* gp = (const float*)&s4[buf][gC];
          float Tl = 1.0f, sr = 0.f, sg = 0.f, sb = 0.f;
          #pragma unroll
          for (int j = 0; j < 8; ++j) {
            // alpha = op*exp(min(power,0)) = exp2(min(q, log2(op)))
            const float q = fmin1(acc[j], gp[4 * j + 0]);
            float al = __builtin_amdgcn_exp2f(q);
            al = fmin1(al, 0.999f);
            al = (al < (1.0f / 255.0f)) ? 0.0f : al;
            const float w = Tl * al;
            sr += w * gp[4 * j + 1];
            sg += w * gp[4 * j + 2];
            sb += w * gp[4 * j + 3];
            Tl -= w;                        // == Tl * (1 - al)
          }
          // stitch halves (gaussians 0-7 before 8-15) across lane pair L,L+16
          const float oT  = __shfl_xor(Tl, 16, 32);
          const float osr = __shfl_xor(sr, 16, 32);
          const float osg = __shfl_xor(sg, 16, 32);
          const float osb = __shfl_xor(sb, 16, 32);
          float Plo, slr, slg, slb, shr, shg, shb;
          if (seg == 0) { Plo = Tl; slr = sr;  slg = sg;  slb = sb;
                          shr = osr; shg = osg; shb = osb; }
          else          { Plo = oT; slr = osr; slg = osg; slb = osb;
                          shr = sr;  shg = sg;  shb = sb; }
          ir[r] += T[r] * (slr + Plo * shr);
          ig[r] += T[r] * (slg + Plo * shg);
          ib[r] += T[r] * (slb + Plo * shb);
          T[r]  *= Plo * (seg == 0 ? oT : Tl);   // T *= Pi_lo * Pi_hi
        }
      }
    }
  }

  if (seg == 0) {  // pixel state is duplicated in both halves; low half writes
    for (int r = 0; r < 2; ++r) {
      const int y = blockIdx.y * 16 + 2 * wav + r;
      const int x = blockIdx.x * 16 + colN;
      const long pix = (long)b * (IMG * IMG) + (long)y * IMG + x;
      out[pix * 3 + 0] = ir[r];
      out[pix * 3 + 1] = ig[r];
      out[pix * 3 + 2] = ib[r];
      out[(long)NB * IMG * IMG * 3 + pix] = 1.0f - T[r];
    }
  }
}

extern "C" void kernel_launch(void* const* d_in, const int* in_sizes, int n_in,
                              void* d_out, int out_size, void* d_ws, size_t ws_size,
                              hipStream_t stream) {
  const float* means2d = (const float*)d_in[0];
  const float* conics  = (const float*)d_in[1];
  const float* colors  = (const float*)d_in[2];
  const float* opac    = (const float*)d_in[3];
  const float* depths  = (const float*)d_in[4];
  int* order = (int*)d_ws;  // NB*GTOT ints = 64 KB

  gs_sort_rank<<<dim3(GTOT / 256, NB), 256, 0, stream>>>(depths, order);
  gs_raster<<<dim3(IMG / 16, IMG / 16, NB), 256, 0, stream>>>(
      means2d, conics, colors, opac, order, (float*)d_out);
}